// TabularDecoder_84482006712641
// MI455X (gfx1250) — compile-verified
//
#include <hip/hip_runtime.h>

#define DIMS 128
#define NCLS 32
#define NCOL 8

typedef float v2f __attribute__((ext_vector_type(2)));
typedef float v4f __attribute__((ext_vector_type(4)));
typedef float v8f __attribute__((ext_vector_type(8)));

// ---------------- routing: histogram + prefix + scatter ----------------

__global__ void td_zero_counts(int* ws) {
    if (threadIdx.x < 16) ws[threadIdx.x] = 0;   // counts[8] + cursors[8]
}

__global__ void td_count(const int* __restrict__ vtype, const int* __restrict__ vcol,
                         int n, int* __restrict__ counts) {
    int i = blockIdx.x * blockDim.x + threadIdx.x;
    if (i < n && vtype[i] == 1) atomicAdd(&counts[vcol[i]], 1);
}

__global__ void td_offsets(const int* __restrict__ counts, int* __restrict__ offsets) {
    if (threadIdx.x == 0) {
        int run = 0;
        for (int k = 0; k < NCOL; ++k) { offsets[k] = run; run += counts[k]; }
    }
}

__global__ void td_scatter(const int* __restrict__ vtype, const int* __restrict__ vcol,
                           int n, int* __restrict__ cursors,
                           const int* __restrict__ offsets, int* __restrict__ idxbuf) {
    int i = blockIdx.x * blockDim.x + threadIdx.x;
    if (i < n && vtype[i] == 1) {
        int c = vcol[i];
        int pos = offsets[c] + atomicAdd(&cursors[c], 1);
        idxbuf[pos] = i;
    }
}

// -------- numeric decoder (wave-per-cell) + zero-fill of non-cat rows --------

__global__ void td_num_zero(const float* __restrict__ emb,
                            const int* __restrict__ vtype,
                            const float* __restrict__ wnum,
                            const float* __restrict__ bnum,
                            float* __restrict__ out_num,
                            float* __restrict__ out_cat,
                            int n) {
    int wid  = threadIdx.x >> 5;
    int lane = threadIdx.x & 31;
    int cell = blockIdx.x * (blockDim.x >> 5) + wid;
    if (cell >= n) return;
    int t = vtype[cell];
    v4f e = *(const v4f*)(emb + (size_t)cell * DIMS + lane * 4);
    v4f w = *(const v4f*)(wnum + lane * 4);
    float p = e[0]*w[0] + e[1]*w[1] + e[2]*w[2] + e[3]*w[3];
    p += __shfl_xor(p, 16, 32);
    p += __shfl_xor(p,  8, 32);
    p += __shfl_xor(p,  4, 32);
    p += __shfl_xor(p,  2, 32);
    p += __shfl_xor(p,  1, 32);
    if (lane == 0) out_num[cell] = (t == 0) ? (p + bnum[0]) : 0.0f;
    if (t != 1) out_cat[(size_t)cell * NCLS + lane] = 0.0f;   // routed kernel skips these rows
}

// -------- routed categorical decoder: fp32 WMMA 16x16x4 grouped GEMM --------

__global__ void __launch_bounds__(128)
td_cat_wmma(const float* __restrict__ emb,
            const float* __restrict__ catW,   // [K, C, D]
            const float* __restrict__ catB,   // [K, C]
            const int* __restrict__ counts,
            const int* __restrict__ offsets,
            const int* __restrict__ idxbuf,
            float* __restrict__ out_cat) {
    __shared__ float wlds[NCLS * DIMS];       // 16 KB: this column's full decoder
    const int col = blockIdx.y;

    // cooperative coalesced load of cat_W[col] into LDS
    {
        const v4f* src = (const v4f*)(catW + (size_t)col * NCLS * DIMS);
        v4f* dst = (v4f*)wlds;
        for (int i = threadIdx.x; i < NCLS * DIMS / 4; i += blockDim.x) dst[i] = src[i];
    }
    __syncthreads();

    const int cnt = counts[col];
    const int ntiles = (cnt + 15) >> 4;
    if (cnt == 0) return;
    const int colOff = offsets[col];
    const int wave = threadIdx.x >> 5;
    const int lane = threadIdx.x & 31;
    const int hi   = lane >> 4;               // half-wave select (K split / M split)
    const int n0   = lane & 15;               // class (B/C/D lane = N), also A row m
    const float bias0 = catB[col * NCLS + n0];
    const float bias1 = catB[col * NCLS + n0 + 16];

    const float* brow0 = wlds + n0 * DIMS + 2 * hi;          // B tile, classes 0..15
    const float* brow1 = wlds + (n0 + 16) * DIMS + 2 * hi;   // B tile, classes 16..31

    for (int t = blockIdx.x * 4 + wave; t < ntiles; t += gridDim.x * 4) {
        // gather A: lane supplies row m = n0; tail clamps to last cell
        // (duplicate rows later store identical values -> no EXEC divergence)
        int p = t * 16 + n0;
        if (p > cnt - 1) p = cnt - 1;
        const int cell = idxbuf[colOff + p];
        const float* arow = emb + (size_t)cell * DIMS + 2 * hi;

        v8f acc0 = {};
        v8f acc1 = {};
        #pragma unroll 8
        for (int d0 = 0; d0 < DIMS; d0 += 4) {
            v2f a  = *(const v2f*)(arow + d0);       // A[m][d0 + 2*hi + j]
            v2f b0 = *(const v2f*)(brow0 + d0);      // B[d0 + 2*hi + j][n]
            v2f b1 = *(const v2f*)(brow1 + d0);
            acc0 = __builtin_amdgcn_wmma_f32_16x16x4_f32(
                false, a, false, b0, (short)0, acc0, false, false);
            acc1 = __builtin_amdgcn_wmma_f32_16x16x4_f32(
                false, a, false, b1, (short)0, acc1, false, false);
        }

        // D layout: lane -> N = n0 (+16 for tile1), VGPR r -> M = r + 8*hi
        #pragma unroll
        for (int r = 0; r < 8; ++r) {
            int m  = r + 8 * hi;
            int pm = t * 16 + m;
            if (pm > cnt - 1) pm = cnt - 1;
            const int cm = idxbuf[colOff + pm];
            out_cat[(size_t)cm * NCLS + n0]      = acc0[r] + bias0;
            out_cat[(size_t)cm * NCLS + n0 + 16] = acc1[r] + bias1;
        }
    }
}

extern "C" void kernel_launch(void* const* d_in, const int* in_sizes, int n_in,
                              void* d_out, int out_size, void* d_ws, size_t ws_size,
                              hipStream_t stream) {
    const float* emb  = (const float*)d_in[0];
    const int*   vty  = (const int*)d_in[1];
    const int*   vcl  = (const int*)d_in[2];
    const float* wnum = (const float*)d_in[3];
    const float* bnum = (const float*)d_in[4];
    const float* catW = (const float*)d_in[5];
    const float* catB = (const float*)d_in[6];
    const int n = in_sizes[1];                 // N cells

    float* out_num = (float*)d_out;
    float* out_cat = out_num + n;

    int* ws_i    = (int*)d_ws;
    int* counts  = ws_i;          // 8
    int* cursors = ws_i + 8;      // 8
    int* offsets = ws_i + 16;     // 8
    int* idxbuf  = ws_i + 32;     // N ints (~2 MB)

    const int blocks = (n + 255) / 256;
    td_zero_counts<<<1, 32, 0, stream>>>(ws_i);
    td_count<<<blocks, 256, 0, stream>>>(vty, vcl, n, counts);
    td_offsets<<<1, 32, 0, stream>>>(counts, offsets);
    td_scatter<<<blocks, 256, 0, stream>>>(vty, vcl, n, cursors, offsets, idxbuf);

    td_num_zero<<<(n + 7) / 8, 256, 0, stream>>>(emb, vty, wnum, bnum,
                                                 out_num, out_cat, n);

    dim3 g(128, NCOL);
    td_cat_wmma<<<g, 128, 0, stream>>>(emb, catW, catB, counts, offsets,
                                       idxbuf, out_cat);
}